// GATLayerBatchEnabled_4432406249499
// MI455X (gfx1250) — compile-verified
//
#include <hip/hip_runtime.h>
#include <cstdint>
#include <cstddef>

// ---- problem constants (from reference) ----
#define S_DIM   8
#define N_DIM   2048
#define IN_FEAT 256
#define OUT_FEAT 64
#define SLOPE   0.1f
#define NEG_INF (-9.0e15f)

typedef __attribute__((ext_vector_type(2))) float v2f;
typedef __attribute__((ext_vector_type(8))) float v8f;

__device__ __forceinline__ float leaky(float x) { return x >= 0.0f ? x : SLOPE * x; }

// ---------------------------------------------------------------------------
// Kernel 1: h[s,n,f] = inDoc[s,n,:] . W_w[f,:] + W_b[f]
// One wave computes ONE 16x16 output tile over full K=256 with
// V_WMMA_F32_16X16X4_F32 (4096 waves total -> ~4 waves/SIMD).
// A (16x4 f32): lane l holds row (l&15), K = 2*(l>>4)+{0,1}  -> one float2 load
// B (4x16 f32): lane l holds col (l&15), K = 2*(l>>4)+{0,1}  -> one float2 load
// ---------------------------------------------------------------------------
__global__ void k_hidden(const float* __restrict__ inDoc,
                         const float* __restrict__ Ww,
                         const float* __restrict__ Wb,
                         float* __restrict__ h) {
    const int wave = (blockIdx.x * blockDim.x + threadIdx.x) >> 5;   // 0..4095
    const int lane = threadIdx.x & 31;
    const int tile = wave >> 2;          // 0..1023  (16-row tile)
    const int t    = wave & 3;           // which 16-col slice of the 64 outputs
    const int r  = lane & 15;
    const int kh = lane >> 4;
    const int rowBase = tile * 16;
    const int f = t * 16 + r;

    const float* __restrict__ aRow = inDoc + (size_t)(rowBase + r) * IN_FEAT;
    const float* __restrict__ bRow = Ww + (size_t)f * IN_FEAT;

    v8f acc = (v8f){};
    #pragma unroll 8
    for (int k0 = 0; k0 < IN_FEAT; k0 += 4) {
        const int ka = k0 + 2 * kh;                         // even -> 8B aligned
        const v2f a = *(const v2f*)(aRow + ka);
        const v2f b = *(const v2f*)(bRow + ka);
        acc = __builtin_amdgcn_wmma_f32_16x16x4_f32(
            false, a, false, b, (short)0, acc, false, false);
    }

    // D layout: lane l holds col f; VGPR rr -> row rr + 8*(l>>4)
    const float bias = Wb[f];
    #pragma unroll
    for (int rr = 0; rr < 8; ++rr) {
        const int m = rr + 8 * kh;
        h[(size_t)(rowBase + m) * OUT_FEAT + f] = acc[rr] + bias;
    }
}

// ---------------------------------------------------------------------------
// Kernel 2: score_i[row] = h[row,:] . a1 ;  score_j[row] = h[row,:] . a2
// One thread per row (16384 threads).
// ---------------------------------------------------------------------------
__global__ void k_scores(const float* __restrict__ h,
                         const float* __restrict__ aw,   // (1, 2*OUT_FEAT)
                         float* __restrict__ score_i,
                         float* __restrict__ score_j) {
    const int row = blockIdx.x * blockDim.x + threadIdx.x;   // 0..16383
    const float* __restrict__ hr = h + (size_t)row * OUT_FEAT;
    float si = 0.0f, sj = 0.0f;
    #pragma unroll
    for (int f = 0; f < OUT_FEAT; ++f) {
        const float hv = hr[f];
        si += hv * aw[f];
        sj += hv * aw[OUT_FEAT + f];
    }
    score_i[row] = si;
    score_j[row] = sj;
}

// ---------------------------------------------------------------------------
// Kernel 3: one wave per attention row (s,i):
//  - streams adj[s,i,:] (the ONLY full read of adj in the pipeline)
//  - online softmax max/sum of masked leaky scores
//  - packs validity into a 1-bit/edge mask via __ballot (134MB -> 4MB for pass 2)
// ---------------------------------------------------------------------------
__global__ void k_stats(const int* __restrict__ adj,
                        const float* __restrict__ score_i,
                        const float* __restrict__ score_j,
                        const float* __restrict__ ab_p,
                        float* __restrict__ rowMax,
                        float* __restrict__ rowSum,
                        unsigned* __restrict__ mask) {
    const int wave = (blockIdx.x * blockDim.x + threadIdx.x) >> 5;  // 0..16383
    const int lane = threadIdx.x & 31;
    const int row  = wave;
    const int s    = row >> 11;                                     // row / 2048

    const float si = score_i[row];
    const float ab = ab_p[0];
    const int*   __restrict__ arow = adj + (size_t)row * N_DIM;
    const float* __restrict__ sjp  = score_j + (size_t)s * N_DIM;
    unsigned*    __restrict__ mrow = mask + (size_t)row * (N_DIM / 32);

    float m = -INFINITY, sum = 0.0f;
    for (int chunk = 0; chunk < N_DIM / 32; ++chunk) {
        const int j = chunk * 32 + lane;
        const int a = arow[j];
        const bool valid = !(a == 0 || a == -1);
        float v = leaky(si + sjp[j] + ab);
        if (!valid) v = NEG_INF;
        if (v > m) { sum *= __expf(m - v); m = v; }
        sum += __expf(v - m);
        const unsigned bits = (unsigned)__ballot(valid);  // bit position == lane == j&31
        if (lane == 0) mrow[chunk] = bits;
    }
    // wave32 reduction of (max, sum) pairs
    #pragma unroll
    for (int off = 16; off >= 1; off >>= 1) {
        const float om = __shfl_xor(m,   off, 32);
        const float os = __shfl_xor(sum, off, 32);
        const float nm = fmaxf(m, om);
        sum = sum * __expf(m - nm) + os * __expf(om - nm);
        m = nm;
    }
    if (lane == 0) { rowMax[row] = m; rowSum[row] = sum; }
}

// ---------------------------------------------------------------------------
// Kernel 4 (fused): one 256-thread block (8 waves) per 16-row i-tile.
// Each wave owns a disjoint 256-wide j slice (K-split):
//   att = exp(masked_leaky(score) - max)/sum   (bitmask + L2-resident stats)
//   -> written once to d_out's att region AND used as the WMMA A operand.
// Partial 16x64 f32 accumulators are reduced through LDS; coalesced h_out
// store with leaky applied. 8192 waves total -> real latency hiding.
// ---------------------------------------------------------------------------
__global__ void __launch_bounds__(256)
k_att_gemm(const float* __restrict__ h,
           const float* __restrict__ score_i,
           const float* __restrict__ score_j,
           const float* __restrict__ ab_p,
           const float* __restrict__ rowMax,
           const float* __restrict__ rowSum,
           const unsigned* __restrict__ mask,
           float* __restrict__ h_out,
           float* __restrict__ att_out) {
    __shared__ float red[8 * 16 * OUT_FEAT];               // 32 KB

    const int s     = blockIdx.x >> 7;                     // blockIdx.x / 128
    const int iBase = (blockIdx.x & 127) * 16;
    const int w     = threadIdx.x >> 5;                    // wave in block: 0..7
    const int lane  = threadIdx.x & 31;
    const int r  = lane & 15;
    const int kh = lane >> 4;

    const size_t gRow = (size_t)s * N_DIM + (iBase + r);   // this lane's att row
    const float si   = score_i[gRow];
    const float ab   = ab_p[0];
    const float M    = rowMax[gRow];
    const float invS = 1.0f / rowSum[gRow];

    const float*    __restrict__ sjp    = score_j + (size_t)s * N_DIM;
    const unsigned* __restrict__ mrow   = mask + gRow * (N_DIM / 32);
    float*          __restrict__ attRow = att_out + gRow * N_DIM;
    const float*    __restrict__ hS     = h + (size_t)s * N_DIM * OUT_FEAT;

    v8f acc[4];
    #pragma unroll
    for (int t = 0; t < 4; ++t) acc[t] = (v8f){};

    // this wave's j slice: blocks [w*8, w*8+8) of 32 j's each -> j in [w*256, w*256+256)
    for (int blk = w * 8; blk < w * 8 + 8; ++blk) {
        const unsigned mw = mrow[blk];
        const int base = blk * 32;
        #pragma unroll 2
        for (int kk = 0; kk < 32; kk += 4) {
            const int jb = kk + 2 * kh;                    // bit index (even)
            const int j  = base + jb;
            const v2f sj = *(const v2f*)(sjp + j);
            float v0 = leaky(si + sj.x + ab);
            float v1 = leaky(si + sj.y + ab);
            if (!((mw >> jb) & 1u))       v0 = NEG_INF;
            if (!((mw >> (jb + 1)) & 1u)) v1 = NEG_INF;
            v2f av;
            av.x = __expf(v0 - M) * invS;
            av.y = __expf(v1 - M) * invS;
            *(v2f*)(attRow + j) = av;                      // att output, written once

            const float* __restrict__ bp = hS + (size_t)j * OUT_FEAT + r;
            #pragma unroll
            for (int t = 0; t < 4; ++t) {
                v2f bv;
                bv.x = bp[t * 16];                         // K = j   , col t*16+r
                bv.y = bp[OUT_FEAT + t * 16];              // K = j+1 , col t*16+r
                acc[t] = __builtin_amdgcn_wmma_f32_16x16x4_f32(
                    false, av, false, bv, (short)0, acc[t], false, false);
            }
        }
    }

    // dump partial 16x64 tile to LDS: red[w][row_local][f]
    #pragma unroll
    for (int t = 0; t < 4; ++t) {
        #pragma unroll
        for (int rr = 0; rr < 8; ++rr) {
            const int m = rr + 8 * kh;                     // row within tile
            red[w * (16 * OUT_FEAT) + m * OUT_FEAT + t * 16 + r] = acc[t][rr];
        }
    }
    __syncthreads();

    // reduce 8 partials; tile spans a contiguous 1024-float range of h_out
    float* __restrict__ outTile = h_out + ((size_t)s * N_DIM + iBase) * OUT_FEAT;
    for (int idx = threadIdx.x; idx < 16 * OUT_FEAT; idx += 256) {
        float v = 0.0f;
        #pragma unroll
        for (int ww = 0; ww < 8; ++ww) v += red[ww * (16 * OUT_FEAT) + idx];
        outTile[idx] = leaky(v);
    }
}

// ---------------------------------------------------------------------------
extern "C" void kernel_launch(void* const* d_in, const int* in_sizes, int n_in,
                              void* d_out, int out_size, void* d_ws, size_t ws_size,
                              hipStream_t stream) {
    (void)in_sizes; (void)n_in; (void)out_size; (void)ws_size;

    const float* inDoc = (const float*)d_in[0];   // (S,N,IN_FEAT)
    const int*   adj   = (const int*)  d_in[1];   // (S,N,N,1)
    const float* Ww    = (const float*)d_in[2];   // (OUT_FEAT, IN_FEAT)
    const float* Wb    = (const float*)d_in[3];   // (OUT_FEAT)
    const float* aw    = (const float*)d_in[4];   // (1, 2*OUT_FEAT)
    const float* ab    = (const float*)d_in[5];   // (1)

    float* out     = (float*)d_out;
    float* h_out   = out;                                             // S*N*OUT_FEAT
    float* att_out = out + (size_t)S_DIM * N_DIM * OUT_FEAT;          // S*N*N

    // workspace layout (floats): h | score_i | score_j | rowMax | rowSum | mask(u32)
    float* ws       = (float*)d_ws;
    const size_t ROWS = (size_t)S_DIM * N_DIM;                        // 16384
    float*    h_ws    = ws;
    float*    score_i = ws + ROWS * OUT_FEAT;
    float*    score_j = score_i + ROWS;
    float*    rowMax  = score_j + ROWS;
    float*    rowSum  = rowMax + ROWS;
    unsigned* mask    = (unsigned*)(rowSum + ROWS);                   // ROWS * 64 words

    // 1) h = inDoc @ W^T + b : 4096 waves (one 16x16 tile each)
    k_hidden<<<dim3(1024), dim3(128), 0, stream>>>(inDoc, Ww, Wb, h_ws);
    // 2) per-row attention scores : 1 thread/row
    k_scores<<<dim3(64), dim3(256), 0, stream>>>(h_ws, aw, score_i, score_j);
    // 3) softmax stats + bit-packed adjacency (single full read of adj)
    k_stats<<<dim3(2048), dim3(256), 0, stream>>>(adj, score_i, score_j, ab,
                                                  rowMax, rowSum, mask);
    // 4) fused att materialization + h_out = leaky(att @ h), K-split 8 ways
    k_att_gemm<<<dim3(1024), dim3(256), 0, stream>>>(h_ws, score_i, score_j, ab,
                                                     rowMax, rowSum, mask,
                                                     h_out, att_out);
}